// OntologyNN_29180007809800
// MI455X (gfx1250) — compile-verified
//
#include <hip/hip_runtime.h>

// ---------------------------------------------------------------------------
// OntologyNN tree evaluation, MI455X (gfx1250).
// Memory-bound (0.25 flop/byte): 614 MB traffic -> ~26us floor @ 23.3 TB/s.
// - CDNA5 async global->LDS (ASYNCcnt) stages each leaf byte from HBM once.
// - NT load/store policies keep the 192MB L2 free for the shared 150KB
//   sigmoid(w) table that every workgroup re-reads.
// - Tree reduction (4681 internal nodes/row) runs entirely in LDS.
// ---------------------------------------------------------------------------

#define TPB 256
#define NUM_NODES 37449           // (8^6-1)/7
#define NUM_INTERNAL 4681         // nodes 0..4680
#define LEAF_START 4681
#define NUM_LEAVES 32768
#define L4_START 585              // level-4 parents: 585..4680 (4096)
#define CHUNK_LEAVES 8192         // leaves staged per chunk (32 KB)
#define CHUNK_VEC4 (CHUNK_LEAVES / 4)        // 2048 float4 tiles per chunk
#define NCHUNK (NUM_LEAVES / CHUNK_LEAVES)   // 4
#define PARENTS_PER_CHUNK (CHUNK_LEAVES / 8) // 1024 level-4 parents per chunk

typedef float v4f __attribute__((ext_vector_type(4)));

__device__ __forceinline__ float sigmoidf_dev(float x) {
    return 1.0f / (1.0f + expf(-x));
}

__device__ __forceinline__ float clamp01(float x) {
    return fminf(fmaxf(x, 0.0f), 1.0f);
}

// Raw LDS byte offset of a generic pointer that points into shared memory.
__device__ __forceinline__ unsigned lds_off_u32(const void* p) {
    return (unsigned)(unsigned long long)(__attribute__((address_space(3))) const void*)p;
}

// CDNA5 async copy: 16B per lane, global -> LDS, tracked by ASYNCcnt.
// NT temporal hint: leaf data is touched exactly once -> don't cache it.
__device__ __forceinline__ void async_g2lds_b128(unsigned lds_byte_addr, const void* gaddr) {
    asm volatile("global_load_async_to_lds_b128 %0, %1, off th:TH_LOAD_NT"
                 :
                 : "v"(lds_byte_addr), "v"(gaddr)
                 : "memory");
}

__device__ __forceinline__ void wait_asynccnt0() {
    asm volatile("s_wait_asynccnt 0x0" ::: "memory");
}

// Weighted sum of 8 contiguous children, clamped to [0,1].
template <bool USE_SW>
__device__ __forceinline__ float wsum8(const float* __restrict__ c,
                                       const float* __restrict__ w) {
    float s = 0.0f;
#pragma unroll
    for (int k = 0; k < 8; ++k) {
        float wk = USE_SW ? w[k] : sigmoidf_dev(w[k]);
        s = fmaf(wk, c[k], s);
    }
    return clamp01(s);
}

__global__ void sigmoid_kernel(const float* __restrict__ w,
                               float* __restrict__ sw, int n) {
    int i = blockIdx.x * blockDim.x + threadIdx.x;
    if (i < n) sw[i] = 1.0f / (1.0f + expf(-w[i]));
}

// One workgroup (8 waves) per batch row.
template <bool USE_SW>
__global__ __launch_bounds__(TPB) void tree_kernel(const float* __restrict__ in,
                                                   const float* __restrict__ wsrc,
                                                   float* __restrict__ out) {
    __shared__ __align__(16) float s_leaf[CHUNK_LEAVES + 8]; // staged leaf window
    __shared__ float s_node[NUM_INTERNAL];                   // computed internal nodes

    const int row = blockIdx.x;
    const int t   = threadIdx.x;
    const float* rin  = in  + (size_t)row * NUM_NODES;
    float*       rout = out + (size_t)row * NUM_NODES;

    // Row stride 37449 is odd -> leaf start is not 16B aligned for most rows.
    // ef = #floats to step back so the staging window start is 16B aligned.
    const int ef = (int)(((unsigned)((size_t)(rin + LEAF_START) & 15u)) >> 2);

    const unsigned lds_base = lds_off_u32(s_leaf); // 16B aligned
    const v4f* s4 = (const v4f*)s_leaf;

    for (int c = 0; c < NCHUNK; ++c) {
        // Staging window: nodes [LEAF_START + c*CHUNK_LEAVES - ef, +CHUNK_LEAVES)
        const int wstart = LEAF_START + c * CHUNK_LEAVES - ef;
        const char* gbase = (const char*)(rin + wstart); // 16B aligned

        // ---- async stage bulk: 2048 x b128 (8 per thread) ----
        for (int j = t; j < CHUNK_VEC4; j += TPB) {
            async_g2lds_b128(lds_base + (unsigned)(j * 16), gbase + (size_t)j * 16);
        }
        // ---- scalar stage the ef tail leaves just past the window ----
        if (t < ef) {
            s_leaf[CHUNK_LEAVES + t] = rin[wstart + CHUNK_LEAVES + t];
        }
        wait_asynccnt0();
        __syncthreads();

        // ---- pass-through copy of leaves (b128 non-temporal stores) ----
        {
            v4f* gout4 = (v4f*)(rout + wstart); // same alignment as input
            const bool skip0 = (c == 0) && (ef > 0); // tile 0 holds ef internal floats
            for (int j = t; j < CHUNK_VEC4; j += TPB) {
                if (skip0 && j == 0) continue;
                __builtin_nontemporal_store(s4[j], &gout4[j]);
            }
            if (c == 0 && ef > 0 && t < 4 - ef) {
                rout[LEAF_START + t] = s_leaf[ef + t];              // leaves in tile 0
            }
            if (c == NCHUNK - 1 && t < ef) {
                rout[NUM_NODES - ef + t] = s_leaf[CHUNK_LEAVES + t]; // row tail
            }
        }

        // ---- level 4: 1024 parents from this chunk's leaves ----
        for (int j = t; j < PARENTS_PER_CHUNK; j += TPB) {
            const int p = L4_START + c * PARENTS_PER_CHUNK + j;
            // children (8p+1..8p+8) -> local leaf index 8*j + ef
            s_node[p] = wsum8<USE_SW>(&s_leaf[8 * j + ef], &wsrc[8 * p + 1]);
        }
        __syncthreads(); // protect s_leaf before next chunk restages it
    }

    // ---- level 3: parents 73..584 ----
    for (int j = t; j < 512; j += TPB) {
        const int p = 73 + j;
        s_node[p] = wsum8<USE_SW>(&s_node[8 * p + 1], &wsrc[8 * p + 1]);
    }
    __syncthreads();

    // ---- level 2: parents 9..72 ----
    if (t < 64) {
        const int p = 9 + t;
        s_node[p] = wsum8<USE_SW>(&s_node[8 * p + 1], &wsrc[8 * p + 1]);
    }
    __syncthreads();

    // ---- level 1: parents 1..8 ----
    if (t < 8) {
        const int p = 1 + t;
        s_node[p] = wsum8<USE_SW>(&s_node[8 * p + 1], &wsrc[8 * p + 1]);
    }
    __syncthreads();

    // ---- level 0: root ----
    if (t == 0) {
        s_node[0] = wsum8<USE_SW>(&s_node[1], &wsrc[1]);
    }
    __syncthreads();

    // ---- write computed internal nodes (coalesced, non-temporal) ----
    for (int i = t; i < NUM_INTERNAL; i += TPB) {
        __builtin_nontemporal_store(s_node[i], &rout[i]);
    }
}

extern "C" void kernel_launch(void* const* d_in, const int* in_sizes, int n_in,
                              void* d_out, int out_size, void* d_ws, size_t ws_size,
                              hipStream_t stream) {
    (void)in_sizes; (void)n_in; (void)out_size;
    const float* probs = (const float*)d_in[0];
    const float* w     = (const float*)d_in[1];
    float* out = (float*)d_out;

    const int batch = 2048;
    const bool use_ws = (d_ws != nullptr) && (ws_size >= (size_t)NUM_NODES * sizeof(float));

    if (use_ws) {
        float* sw = (float*)d_ws;
        sigmoid_kernel<<<(NUM_NODES + 255) / 256, 256, 0, stream>>>(w, sw, NUM_NODES);
        tree_kernel<true><<<batch, TPB, 0, stream>>>(probs, sw, out);
    } else {
        // Fallback: compute sigmoid(w) on the fly inside the tree kernel.
        tree_kernel<false><<<batch, TPB, 0, stream>>>(probs, w, out);
    }
}